// StructuralAttentionLayer_25580825215754
// MI455X (gfx1250) — compile-verified
//
#include <hip/hip_runtime.h>

// ---------------------------------------------------------------------------
// GATConv forward for MI455X (gfx1250, wave32).
//   xp = x @ W            : fp32 WMMA 16x16x4, one wave per 16x16 tile
//   a_src/a_dst           : per (node,head) 64-wide dots
//   softmax over dst      : atomic flipped-uint max, exp, atomic f32 sum
//   out[dst] += coef*xp[src] : native f32 atomics (L2-resident, 192MB L2)
// ---------------------------------------------------------------------------

typedef float v2f __attribute__((ext_vector_type(2)));
typedef float v8f __attribute__((ext_vector_type(8)));

#define IN_CH   128
#define HC      256   // HEADS * OUT_CH
#define NEG_SLOPE 0.2f

// ---- monotone float <-> uint mapping for atomic max on floats -------------
__device__ __forceinline__ unsigned flipf(float f) {
  unsigned u = __float_as_uint(f);
  return (u & 0x80000000u) ? ~u : (u | 0x80000000u);
}
__device__ __forceinline__ float unflipf(unsigned u) {
  return __uint_as_float((u & 0x80000000u) ? (u ^ 0x80000000u) : ~u);
}

// ---- init: amax=0 (== -inf in flipped space), denom=0, out=bias -----------
__global__ void gat_init(unsigned* __restrict__ amax, float* __restrict__ denom,
                         float* __restrict__ out, const float* __restrict__ bias,
                         int N) {
  int stride = gridDim.x * blockDim.x;
  for (int i = blockIdx.x * blockDim.x + threadIdx.x; i < N * 4; i += stride) {
    amax[i]  = 0u;
    denom[i] = 0.0f;
  }
  // out = bias (broadcast over rows), vectorized: N*64 float4 stores
  float4* o4 = (float4*)out;
  const float4* b4 = (const float4*)bias;
  int total4 = N * (HC / 4);
  for (int j = blockIdx.x * blockDim.x + threadIdx.x; j < total4; j += stride) {
    o4[j] = b4[j & (HC / 4 - 1)];
  }
}

// ---- xp = x @ W via V_WMMA_F32_16X16X4_F32 --------------------------------
// A (16x4 f32): lane m=lane&15, koff=(lane>>4)*2 -> {A[m][k+koff], A[m][k+koff+1]}
// B (4x16 f32): lane n=lane&15, same koff        -> {B[k+koff][n], B[k+koff+1][n]}
// C/D (16x16)  : 8 VGPRs, element r -> row (r + 8*(lane>>4)), col (lane&15)
__global__ __launch_bounds__(256)
void gat_gemm_xp(const float* __restrict__ x, const float* __restrict__ W,
                 float* __restrict__ xp, int N, int tilesM) {
  const int tilesN = HC / 16;                       // 16
  int wave = blockIdx.x * (blockDim.x / 32) + (threadIdx.x >> 5);
  int lane = threadIdx.x & 31;
  int tileM = wave / tilesN;
  int tileN = wave % tilesN;
  if (tileM >= tilesM) return;                      // wave-uniform exit

  int m    = tileM * 16 + (lane & 15);
  int n    = tileN * 16 + (lane & 15);
  int koff = (lane >> 4) * 2;
  int mc   = m < N ? m : (N - 1);                   // clamp loads (EXEC stays full)

  const float* xrow = x + (size_t)mc * IN_CH;
  v8f acc = {};
  #pragma unroll 4
  for (int k = 0; k < IN_CH; k += 4) {
    v2f a, b;
    a.x = xrow[k + koff];
    a.y = xrow[k + koff + 1];
    b.x = W[(size_t)(k + koff)     * HC + n];
    b.y = W[(size_t)(k + koff + 1) * HC + n];
    acc = __builtin_amdgcn_wmma_f32_16x16x4_f32(false, a, false, b,
                                                (short)0, acc, false, false);
  }

  int row0 = tileM * 16 + (lane >> 4) * 8;
  int col  = tileN * 16 + (lane & 15);
  float* dst = xp + (size_t)row0 * HC + col;
  if (tileM * 16 + 16 <= N) {                       // uniform fast path
    #pragma unroll
    for (int r = 0; r < 8; ++r) dst[(size_t)r * HC] = acc[r];
  } else {                                          // rare ragged last tile
    #pragma unroll
    for (int r = 0; r < 8; ++r)
      if (row0 + r < N) dst[(size_t)r * HC] = acc[r];
  }
}

// ---- a_src[n,h], a_dst[n,h] ----------------------------------------------
__global__ void gat_att(const float* __restrict__ xp,
                        const float* __restrict__ att_src,
                        const float* __restrict__ att_dst,
                        float* __restrict__ a_src, float* __restrict__ a_dst,
                        int N) {
  int idx = blockIdx.x * blockDim.x + threadIdx.x;   // (n,h)
  if (idx >= N * 4) return;
  int n = idx >> 2, h = idx & 3;
  const float* row = xp + (size_t)n * HC + h * 64;
  const float* as  = att_src + h * 64;
  const float* ad  = att_dst + h * 64;
  float s = 0.0f, d = 0.0f;
  #pragma unroll 4
  for (int c = 0; c < 64; c += 4) {
    float4 v = *(const float4*)(row + c);
    float4 a = *(const float4*)(as + c);
    float4 b = *(const float4*)(ad + c);
    s += v.x * a.x + v.y * a.y + v.z * a.z + v.w * a.w;
    d += v.x * b.x + v.y * b.y + v.z * b.z + v.w * b.w;
  }
  a_src[idx] = s;
  a_dst[idx] = d;
}

// ---- alpha = leaky_relu(a_src[src]+a_dst[dst]); atomic max per (dst,h) ----
__global__ void gat_alpha_max(const int* __restrict__ ei,
                              const float* __restrict__ a_src,
                              const float* __restrict__ a_dst,
                              float* __restrict__ alpha,
                              unsigned* __restrict__ amax, int E, int N) {
  int Etot = E + N;
  int idx = blockIdx.x * blockDim.x + threadIdx.x;   // (e,h)
  if (idx >= Etot * 4) return;
  int e = idx >> 2, h = idx & 3;
  int s = (e < E) ? ei[e]     : (e - E);
  int d = (e < E) ? ei[E + e] : (e - E);
  float al = a_src[s * 4 + h] + a_dst[d * 4 + h];
  al = (al > 0.0f) ? al : NEG_SLOPE * al;
  alpha[idx] = al;
  atomicMax(&amax[d * 4 + h], flipf(al));
}

// ---- ex = exp(alpha - amax[dst]); denom[dst,h] += ex ----------------------
__global__ void gat_expsum(const int* __restrict__ ei,
                           float* __restrict__ alpha,
                           const unsigned* __restrict__ amax,
                           float* __restrict__ denom, int E, int N) {
  int Etot = E + N;
  int idx = blockIdx.x * blockDim.x + threadIdx.x;
  if (idx >= Etot * 4) return;
  int e = idx >> 2, h = idx & 3;
  int d = (e < E) ? ei[E + e] : (e - E);
  float ex = __expf(alpha[idx] - unflipf(amax[d * 4 + h]));
  alpha[idx] = ex;                                   // reuse buffer as ex
  __hip_atomic_fetch_add(&denom[d * 4 + h], ex,
                         __ATOMIC_RELAXED, __HIP_MEMORY_SCOPE_AGENT);
}

// ---- out[dst] += (ex/denom[dst]) * xp[src]; 64 lanes per edge -------------
__global__ __launch_bounds__(256)
void gat_scatter(const int* __restrict__ ei, const float* __restrict__ xp,
                 const float* __restrict__ ex, const float* __restrict__ denom,
                 float* __restrict__ out, int E, int N) {
  int Etot = E + N;
  int lane = threadIdx.x & 63;                       // 4 channels each
  int e    = blockIdx.x * 4 + (threadIdx.x >> 6);
  if (e >= Etot) return;
  int s = (e < E) ? ei[e]     : (e - E);
  int d = (e < E) ? ei[E + e] : (e - E);
  int h = lane >> 4;                                 // channel group -> head
  float coef = ex[e * 4 + h] / denom[d * 4 + h];
  float4 v = *(const float4*)(xp + (size_t)s * HC + lane * 4);
  float* o = out + (size_t)d * HC + lane * 4;
  __hip_atomic_fetch_add(o + 0, coef * v.x, __ATOMIC_RELAXED, __HIP_MEMORY_SCOPE_AGENT);
  __hip_atomic_fetch_add(o + 1, coef * v.y, __ATOMIC_RELAXED, __HIP_MEMORY_SCOPE_AGENT);
  __hip_atomic_fetch_add(o + 2, coef * v.z, __ATOMIC_RELAXED, __HIP_MEMORY_SCOPE_AGENT);
  __hip_atomic_fetch_add(o + 3, coef * v.w, __ATOMIC_RELAXED, __HIP_MEMORY_SCOPE_AGENT);
}

// ---------------------------------------------------------------------------
extern "C" void kernel_launch(void* const* d_in, const int* in_sizes, int n_in,
                              void* d_out, int out_size, void* d_ws, size_t ws_size,
                              hipStream_t stream) {
  const float* x       = (const float*)d_in[0];
  const float* W       = (const float*)d_in[1];
  const float* att_src = (const float*)d_in[2];
  const float* att_dst = (const float*)d_in[3];
  const float* bias    = (const float*)d_in[4];
  const int*   ei      = (const int*)d_in[5];

  int N    = in_sizes[0] / IN_CH;     // 50000
  int E    = in_sizes[5] / 2;         // 800000
  int Etot = E + N;

  // workspace layout (fp32 elements); total ~67 MB
  char* ws = (char*)d_ws;
  float*    xp    = (float*)ws;    ws += (size_t)N * HC * sizeof(float);
  float*    asrc  = (float*)ws;    ws += (size_t)N * 4 * sizeof(float);
  float*    adst  = (float*)ws;    ws += (size_t)N * 4 * sizeof(float);
  unsigned* amax  = (unsigned*)ws; ws += (size_t)N * 4 * sizeof(unsigned);
  float*    denom = (float*)ws;    ws += (size_t)N * 4 * sizeof(float);
  float*    alpha = (float*)ws;    ws += (size_t)Etot * 4 * sizeof(float);
  float*    out   = (float*)d_out;

  gat_init<<<1024, 256, 0, stream>>>(amax, denom, out, bias, N);

  int tilesM = (N + 15) / 16;
  int waves  = tilesM * (HC / 16);                   // one wave per 16x16 tile
  gat_gemm_xp<<<(waves + 7) / 8, 256, 0, stream>>>(x, W, xp, N, tilesM);

  gat_att<<<(N * 4 + 255) / 256, 256, 0, stream>>>(xp, att_src, att_dst, asrc, adst, N);

  gat_alpha_max<<<(Etot * 4 + 255) / 256, 256, 0, stream>>>(ei, asrc, adst, alpha, amax, E, N);

  gat_expsum<<<(Etot * 4 + 255) / 256, 256, 0, stream>>>(ei, alpha, amax, denom, E, N);

  gat_scatter<<<(Etot + 3) / 4, 256, 0, stream>>>(ei, xp, alpha, denom, out, E, N);
}